// NeuralNetwork_30863634989111
// MI455X (gfx1250) — compile-verified
//
#include <hip/hip_runtime.h>

typedef __attribute__((ext_vector_type(2))) float v2f;
typedef __attribute__((ext_vector_type(8))) float v8f;

#define LATENT   512
#define H1       128
#define H2       256
#define NOUT      64
// per-sample parameter layout inside hypernet row (fp32 elements)
#define W1_OFF        0
#define B1_OFF    65536
#define W2_OFF    65664
#define B2_OFF    98432
#define W3_OFF    98688
#define B3_OFF   115072
#define TOTAL_P  115136

// D = A(16x4) * B(4x16) + C, fp32 WMMA. B columns all carry the same x-chunk,
// so every column of C/D is the exact fp32 dot product of its row of A.
__device__ __forceinline__ v8f wmma_acc(v8f c, v2f a, v2f b) {
  return __builtin_amdgcn_wmma_f32_16x16x4_f32(false, a, false, b,
                                               (short)0, c, false, false);
}

__global__ __launch_bounds__(256) void hyper_mlp_kernel(
    const float* __restrict__ xg,      // [B, 512]
    const float* __restrict__ hyper,   // [B, 115136]
    float* __restrict__ outg)          // [B, 64]
{
  __shared__ float sx[LATENT];
  __shared__ float sh1[H1];
  __shared__ float sh2[H2];

  const int b    = blockIdx.x;
  const float* hyp = hyper + (size_t)b * TOTAL_P;
  const int tid  = threadIdx.x;
  const int lane = tid & 31;
  const int wv   = tid >> 5;    // 8 waves per block
  const int row  = lane & 15;   // A-matrix M index for this lane
  const int kh   = lane >> 4;   // 0: K=0,1  1: K=2,3  (A/B VGPR layout halves)

  // stage x[b] into LDS (coalesced: 2 floats per thread)
  {
    const float* xp = xg + (size_t)b * LATENT;
    sx[tid]       = xp[tid];
    sx[tid + 256] = xp[tid + 256];
  }
  __syncthreads();

  // ---------- layer 1: h1 = relu(W1 @ x + b1),  W1: 128x512 ----------
  {
    const int r0 = 16 * wv;                               // 8 waves * 16 rows = 128
    const float* arow = hyp + W1_OFF + (size_t)(r0 + row) * LATENT + 2 * kh;
    v8f c0 = {}, c1 = {};
    for (int k = 0; k < LATENT; k += 8) {
      v2f a0 = *(const v2f*)(arow + k);
      v2f b0; b0.x = sx[k + 2 * kh]; b0.y = sx[k + 2 * kh + 1];
      c0 = wmma_acc(c0, a0, b0);
      v2f a1 = *(const v2f*)(arow + k + 4);
      v2f b1v; b1v.x = sx[k + 4 + 2 * kh]; b1v.y = sx[k + 4 + 2 * kh + 1];
      c1 = wmma_acc(c1, a1, b1v);
    }
    if (row == 0) {                                       // lanes 0 and 16
      const int rbase = r0 + kh * 8;                      // rows j / j+8 per C layout
      const float* bias = hyp + B1_OFF;
#pragma unroll
      for (int j = 0; j < 8; ++j)
        sh1[rbase + j] = fmaxf(c0[j] + c1[j] + bias[rbase + j], 0.0f);
    }
  }
  __syncthreads();

  // ---------- layer 2: h2 = relu(W2 @ h1 + b2),  W2: 256x128 ----------
  {
    const float* w2   = hyp + W2_OFF;
    const float* bias = hyp + B2_OFF;
#pragma unroll
    for (int g = 0; g < 2; ++g) {                         // 8 waves * 2 * 16 rows = 256
      const int r0 = 32 * wv + 16 * g;
      const float* arow = w2 + (size_t)(r0 + row) * H1 + 2 * kh;
      v8f c0 = {}, c1 = {};
      for (int k = 0; k < H1; k += 8) {
        v2f a0 = *(const v2f*)(arow + k);
        v2f b0; b0.x = sh1[k + 2 * kh]; b0.y = sh1[k + 2 * kh + 1];
        c0 = wmma_acc(c0, a0, b0);
        v2f a1 = *(const v2f*)(arow + k + 4);
        v2f b1v; b1v.x = sh1[k + 4 + 2 * kh]; b1v.y = sh1[k + 4 + 2 * kh + 1];
        c1 = wmma_acc(c1, a1, b1v);
      }
      if (row == 0) {
        const int rbase = r0 + kh * 8;
#pragma unroll
        for (int j = 0; j < 8; ++j)
          sh2[rbase + j] = fmaxf(c0[j] + c1[j] + bias[rbase + j], 0.0f);
      }
    }
  }
  __syncthreads();

  // ---------- layer 3: out = W3 @ h2 + b3,  W3: 64x256 (waves 0-3) ----------
  if (wv < 4) {
    const float* w3   = hyp + W3_OFF;
    const float* bias = hyp + B3_OFF;
    const int r0 = 16 * wv;                               // 4 waves * 16 rows = 64
    const float* arow = w3 + (size_t)(r0 + row) * H2 + 2 * kh;
    v8f c0 = {}, c1 = {};
    for (int k = 0; k < H2; k += 8) {
      v2f a0 = *(const v2f*)(arow + k);
      v2f b0; b0.x = sh2[k + 2 * kh]; b0.y = sh2[k + 2 * kh + 1];
      c0 = wmma_acc(c0, a0, b0);
      v2f a1 = *(const v2f*)(arow + k + 4);
      v2f b1v; b1v.x = sh2[k + 4 + 2 * kh]; b1v.y = sh2[k + 4 + 2 * kh + 1];
      c1 = wmma_acc(c1, a1, b1v);
    }
    if (row == 0) {
      const int rbase = r0 + kh * 8;
      float* op = outg + (size_t)b * NOUT;
#pragma unroll
      for (int j = 0; j < 8; ++j)
        op[rbase + j] = c0[j] + c1[j] + bias[rbase + j];
    }
  }
}

extern "C" void kernel_launch(void* const* d_in, const int* in_sizes, int n_in,
                              void* d_out, int out_size, void* d_ws, size_t ws_size,
                              hipStream_t stream) {
  const float* x   = (const float*)d_in[0];   // [B, 512] fp32
  const float* hyp = (const float*)d_in[1];   // [B, 115136] fp32
  float* out = (float*)d_out;                 // [B, 64] fp32
  const int batch = in_sizes[0] / LATENT;     // 2048
  hyper_mlp_kernel<<<batch, 256, 0, stream>>>(x, hyp, out);
}